// SparseAutoencoder_33655363731968
// MI455X (gfx1250) — compile-verified
//
#include <hip/hip_runtime.h>
#include <hip/hip_bf16.h>

// Problem sizes (fixed by the reference)
#define BB 4096
#define DD 1024
#define HH 16384
#define KK 32

typedef __attribute__((ext_vector_type(2))) float v2f;
typedef __attribute__((ext_vector_type(4))) float v4f;
typedef __attribute__((ext_vector_type(8))) float v8f;

// LDS tile geometry (floats). Stride 36: 16B-aligned rows + conflict-free frag reads.
#define A_STRIDE 36
#define A_TILE (128 * A_STRIDE)
#define W_TILE (64 * A_STRIDE)
#define BUF_FLOATS (A_TILE + W_TILE)

// ---------------------------------------------------------------------------
// Kernel 0: corr[h] = dot(bias, W_enc[h,:])  -- folds the bias into a
// batch-independent epilogue term so the GEMM tiles can be raw async DMA.
// ---------------------------------------------------------------------------
__global__ __launch_bounds__(256) void corr_kernel(
    const float* __restrict__ W_enc, const float* __restrict__ bias,
    float* __restrict__ corr) {
  int h = blockIdx.x * 256 + threadIdx.x;
  const float* wr = W_enc + (size_t)h * DD;
  float s = 0.f;
  for (int d = 0; d < DD; d += 4) {
    v4f w = *(const v4f*)(wr + d);
    v4f b = *(const v4f*)(bias + d);
    s += w.x * b.x + w.y * b.y + w.z * b.z + w.w * b.w;
  }
  corr[h] = s;
}

// ---------------------------------------------------------------------------
// Kernel 1: encode GEMM  z = x @ W_enc^T - corr   via V_WMMA_F32_16X16X4_F32
//   block tile 128x64, 8 waves x (32x32 = 2x2 WMMA accumulators)
//   K-chunks of 32 double-buffered in LDS via GLOBAL_LOAD_ASYNC_TO_LDS_B128
//   (ASYNCcnt-tracked, VGPR-bypassing copies overlapped with WMMA compute)
// ---------------------------------------------------------------------------
__global__ __launch_bounds__(256) void encode_wmma_kernel(
    const float* __restrict__ x, const float* __restrict__ W_enc,
    const float* __restrict__ corr, float* __restrict__ z /* [B,H] */) {
  __shared__ __align__(16) float smem[2 * BUF_FLOATS];

  const int tid  = threadIdx.x;
  const int lane = tid & 31;
  const int wave = tid >> 5;   // 0..7
  const int wm   = wave >> 1;  // 0..3 (M)
  const int wn   = wave & 1;   // 0..1 (N)
  const int half = lane >> 4;
  const int lm   = lane & 15;

  const int rowBase = blockIdx.y * 128;  // over B
  const int colBase = blockIdx.x * 64;   // over H

  // numeric LDS byte offset of smem (flat addr low 32 bits == LDS offset)
  const unsigned smemBase = (unsigned)(size_t)&smem[0];

  // Issue one K-chunk's async copies: 4 A-tile + 2 W-tile b128 per thread
  // => 6 ASYNCcnt increments per wave per chunk.
  auto issue = [&](int buf, int k0) {
    const unsigned base = smemBase + (unsigned)(buf * (BUF_FLOATS * 4));
#pragma unroll
    for (int i = 0; i < 4; ++i) {
      int q = tid + (i << 8);
      int r = q >> 3, c4 = (q & 7) << 2;
      const float* g = x + (size_t)(rowBase + r) * DD + k0 + c4;
      unsigned l = base + (unsigned)((r * A_STRIDE + c4) * 4);
      asm volatile("global_load_async_to_lds_b128 %0, %1, off" ::"v"(l), "v"(g)
                   : "memory");
    }
#pragma unroll
    for (int i = 0; i < 2; ++i) {
      int q = tid + (i << 8);
      int r = q >> 3, c4 = (q & 7) << 2;
      const float* g = W_enc + (size_t)(colBase + r) * DD + k0 + c4;
      unsigned l = base + (unsigned)((A_TILE + r * A_STRIDE + c4) * 4);
      asm volatile("global_load_async_to_lds_b128 %0, %1, off" ::"v"(l), "v"(g)
                   : "memory");
    }
  };

  v8f acc[2][2] = {};

  issue(0, 0);
  for (int c = 0; c < DD / 32; ++c) {
    if (c + 1 < DD / 32) {
      issue((c + 1) & 1, (c + 1) * 32);
      // 6 outstanding belong to chunk c+1; <=6 means chunk c has landed
      asm volatile("s_wait_asynccnt 0x6" ::: "memory");
    } else {
      asm volatile("s_wait_asynccnt 0x0" ::: "memory");
    }
    __syncthreads();  // all waves' copies for chunk c now visible

    const float* aS = smem + (c & 1) * BUF_FLOATS;
    const float* wS = aS + A_TILE;

#pragma unroll
    for (int kk = 0; kk < 32; kk += 4) {
      // A frag (16x4 fp32): VGPR0 K=kk+2*half, VGPR1 K=kk+2*half+1, M=lane%16
      // B frag (4x16 fp32): same K striping, N=lane%16 (B[k][n] = W[n][k])
      v2f afr[2], bfr[2];
#pragma unroll
      for (int s = 0; s < 2; ++s) {
        int m = wm * 32 + s * 16 + lm;
        afr[s].x = aS[m * A_STRIDE + kk + 2 * half];
        afr[s].y = aS[m * A_STRIDE + kk + 2 * half + 1];
        int n = wn * 32 + s * 16 + lm;
        bfr[s].x = wS[n * A_STRIDE + kk + 2 * half];
        bfr[s].y = wS[n * A_STRIDE + kk + 2 * half + 1];
      }
#pragma unroll
      for (int sm = 0; sm < 2; ++sm)
#pragma unroll
        for (int sn = 0; sn < 2; ++sn)
          acc[sm][sn] = __builtin_amdgcn_wmma_f32_16x16x4_f32(
              false, afr[sm], false, bfr[sn], (short)0, acc[sm][sn], false,
              false);
    }
    __syncthreads();  // buffer may be refilled next iteration
  }

  // Epilogue: subtract corr[n]; C layout: VGPR v -> M=v(+8 for hi half), N=lane%16
#pragma unroll
  for (int sm = 0; sm < 2; ++sm)
#pragma unroll
    for (int sn = 0; sn < 2; ++sn) {
      int n = colBase + wn * 32 + sn * 16 + lm;
      float cn = corr[n];
#pragma unroll
      for (int v = 0; v < 8; ++v) {
        int m = rowBase + wm * 32 + sm * 16 + v + 8 * half;
        z[(size_t)m * HH + n] = acc[sm][sn][v] - cn;
      }
    }
}

// ---------------------------------------------------------------------------
// Kernel 2: exact top-32 per row via 4-pass radix select on order-mapped fp32
// ---------------------------------------------------------------------------
__device__ __forceinline__ unsigned ordmap(float f) {
  unsigned u = __float_as_uint(f);
  return (u & 0x80000000u) ? ~u : (u | 0x80000000u);
}

__global__ __launch_bounds__(256) void topk_kernel(
    float* __restrict__ enc /* [B,H] in/out */, float* __restrict__ vals,
    int* __restrict__ idxs) {
  __shared__ unsigned hist[256];
  __shared__ unsigned sPrefix, sRemain, sTie;
  __shared__ int sList;

  const int tid = threadIdx.x;
  const int row = blockIdx.x;
  float* zr = enc + (size_t)row * HH;

  if (tid == 0) {
    sPrefix = 0u;
    sRemain = KK;
  }
  for (int pass = 0; pass < 4; ++pass) {
    hist[tid] = 0u;
    __syncthreads();
    const unsigned pfx = sPrefix;
    const int shift = 24 - 8 * pass;
    for (int i = 0; i < HH / 256; ++i) {
      int h = tid + (i << 8);  // coalesced
      unsigned u = ordmap(zr[h]);
      if (pass == 0 || (u >> (shift + 8)) == pfx)
        atomicAdd(&hist[(u >> shift) & 255u], 1u);
    }
    __syncthreads();
    if (tid == 0) {
      unsigned rem = sRemain, cum = 0u;
      int b = 255;
      for (; b > 0; --b) {
        cum += hist[b];
        if (cum >= rem) break;
      }
      if (cum < rem) cum += hist[0];  // b == 0 fallthrough
      sRemain = rem - (cum - hist[b]);
      sPrefix = (pfx << 8) | (unsigned)b;
    }
    __syncthreads();
  }
  const unsigned thr = sPrefix;      // ordered key of 32nd-largest element
  const unsigned remTies = sRemain;  // how many ==thr elements to keep
  if (tid == 0) {
    sList = 0;
    sTie = 0u;
  }
  __syncthreads();

  for (int i = 0; i < HH / 256; ++i) {
    int h = tid + (i << 8);
    float v = zr[h];
    unsigned u = ordmap(v);
    bool keep = false;
    if (u > thr) {
      keep = true;
    } else if (u == thr) {
      unsigned t = atomicAdd(&sTie, 1u);
      if (t < remTies) keep = true;
    }
    if (keep) {
      int p = atomicAdd(&sList, 1);
      vals[row * KK + p] = v;
      idxs[row * KK + p] = h;
    }
    zr[h] = keep ? v : 0.0f;  // dense encoded output, zeros elsewhere
  }
}

// ---------------------------------------------------------------------------
// Kernel 3: sparse decode  decoded[b,d] = b_dec[d] + sum_j val_j * W_enc[idx_j, d]
//   (W_enc == W_dec^T per setup: coalesced row gathers, L2-resident 64 MB)
// ---------------------------------------------------------------------------
__global__ __launch_bounds__(256) void decode_kernel(
    const float* __restrict__ W_enc, const float* __restrict__ b_dec,
    const float* __restrict__ vals, const int* __restrict__ idxs,
    float* __restrict__ out /* [B,D] */) {
  __shared__ float sv[KK];
  __shared__ int si[KK];
  const int row = blockIdx.x;
  const int tid = threadIdx.x;
  if (tid < KK) {
    sv[tid] = vals[row * KK + tid];
    si[tid] = idxs[row * KK + tid];
  }
  __syncthreads();
#pragma unroll
  for (int c = 0; c < DD / 256; ++c) {
    int d = tid + (c << 8);
    float acc = b_dec[d];
#pragma unroll 8
    for (int j = 0; j < KK; ++j)
      acc += sv[j] * W_enc[(size_t)si[j] * DD + d];
    out[(size_t)row * DD + d] = acc;
  }
}

// ---------------------------------------------------------------------------
extern "C" void kernel_launch(void* const* d_in, const int* in_sizes, int n_in,
                              void* d_out, int out_size, void* d_ws,
                              size_t ws_size, hipStream_t stream) {
  const float* x     = (const float*)d_in[0];  // [B,D]
  const float* W_enc = (const float*)d_in[1];  // [H,D]
  // const float* W_dec = (const float*)d_in[2];  // [D,H] (== W_enc^T)
  const float* b_dec = (const float*)d_in[3];  // [D]
  const float* bias  = (const float*)d_in[4];  // [D]

  float* decoded = (float*)d_out;                    // [B,D]
  float* encoded = (float*)d_out + (size_t)BB * DD;  // [B,H]

  float* vals = (float*)d_ws;                                        // [B,K]
  int* idxs   = (int*)((char*)d_ws + sizeof(float) * (size_t)BB * KK);
  float* corr = (float*)((char*)d_ws + 2 * sizeof(float) * (size_t)BB * KK);

  corr_kernel<<<HH / 256, 256, 0, stream>>>(W_enc, bias, corr);
  dim3 gridE(HH / 64, BB / 128);
  encode_wmma_kernel<<<gridE, 256, 0, stream>>>(x, W_enc, corr, encoded);
  topk_kernel<<<BB, 256, 0, stream>>>(encoded, vals, idxs);
  decode_kernel<<<BB, 256, 0, stream>>>(W_enc, b_dec, vals, idxs, decoded);
}